// GNN_59785944760328
// MI455X (gfx1250) — compile-verified
//
#include <hip/hip_runtime.h>

#define NNODES 50000
#define NPAD   50048   // NNODES rounded up to multiple of 64
#define NEDGES 800000
#define HID    128
#define NG     64
#define BN_EPS 1e-5f

typedef float v2f __attribute__((ext_vector_type(2)));
typedef float v8f __attribute__((ext_vector_type(8)));

__device__ __forceinline__ void atomicAddF(float* p, float v) {
    (void)__hip_atomic_fetch_add(p, v, __ATOMIC_RELAXED, __HIP_MEMORY_SCOPE_AGENT);
}

// ---------------------------------------------------------------- utilities
__global__ void k_fill(float* __restrict__ p, float v, int n) {
    int i = blockIdx.x * blockDim.x + threadIdx.x;
    if (i < n) p[i] = v;
}

__global__ void k_deg_accum(const int* __restrict__ dst, float* __restrict__ deg) {
    int e = blockIdx.x * blockDim.x + threadIdx.x;
    if (e < NEDGES) atomicAddF(&deg[dst[e]], 1.0f);
}

__global__ void k_deg_to_dinv(float* __restrict__ d, int n) {
    int i = blockIdx.x * blockDim.x + threadIdx.x;
    if (i < n) {
        float v = d[i];
        d[i] = (v > 0.0f) ? rsqrtf(v) : 0.0f;
    }
}

// pack x[N,10] -> xp[NPAD,16] zero-padded (removes all K/M guards from layer 1)
__global__ void k_pack_x(const float* __restrict__ x, float* __restrict__ xp) {
    int id = blockIdx.x * blockDim.x + threadIdx.x;   // NPAD*16 ids
    if (id >= NPAD * 16) return;
    int i = id >> 4, c = id & 15;
    float v = 0.0f;
    if (i < NNODES && c < 10) v = x[i * 10 + c];
    xp[id] = v;
}

// pack W1[10,128] -> wp[16,128] zero-padded
__global__ void k_pack_w1(const float* __restrict__ w, float* __restrict__ wp) {
    int id = blockIdx.x * blockDim.x + threadIdx.x;   // 16*128 ids
    if (id >= 16 * HID) return;
    int k = id >> 7;
    wp[id] = (k < 10) ? w[id] : 0.0f;
}

// ---------------------------------------------------------- WMMA fp32 GEMM
// Guard-free; all strides compile-time so every load/store is base+imm-offset.
// M (from grid) must be a multiple of 64; KDIM a multiple of 4.
// One wave computes a 64x16 output block = 4 stacked 16x16 WMMA tiles.
// D[M,NOUT] = A[M,KDIM(lda=LDA)] @ B[KDIM,NOUT] (+bias[col] if bias!=null)
// If AGG!=null: AGG[r,c] = D[r,c]*dinv[r]^2 (GCN self-loop init).
template<int KDIM, int NOUT, int LDA>
__global__ __launch_bounds__(256)
void k_gemm_wmma(const float* __restrict__ A,
                 const float* __restrict__ B,
                 float* __restrict__ D,
                 const float* __restrict__ bias,
                 const float* __restrict__ dinv,
                 float* __restrict__ AGG) {
    constexpr int NTN = NOUT / 16;                    // 16-col tiles
    const int lane = threadIdx.x & 31;
    const int wave = threadIdx.x >> 5;
    const int gid  = blockIdx.x * 8 + wave;           // always < mblks*NTN
    const int mblk = gid / NTN;                       // 64-row block
    const int nt   = gid - mblk * NTN;

    const int half = lane >> 4;                       // A/B: K pair select
    const int r    = lane & 15;
    const int n    = nt * 16 + r;                     // B col / D col

    // hoisted bases; the unrolled loop below uses only constant offsets
    const float* Abase = A + (size_t)(mblk * 64 + r) * LDA + 2 * half;
    const float* Bbase = B + (size_t)(2 * half) * NOUT + n;

    v8f acc0 = {}, acc1 = {}, acc2 = {}, acc3 = {};
    #pragma unroll
    for (int k0 = 0; k0 < KDIM; k0 += 4) {
        v2f b;
        b[0] = Bbase[(size_t)(k0    ) * NOUT];
        b[1] = Bbase[(size_t)(k0 + 1) * NOUT];
        v2f a0 = *reinterpret_cast<const v2f*>(Abase            + k0);
        v2f a1 = *reinterpret_cast<const v2f*>(Abase + 16 * LDA + k0);
        v2f a2 = *reinterpret_cast<const v2f*>(Abase + 32 * LDA + k0);
        v2f a3 = *reinterpret_cast<const v2f*>(Abase + 48 * LDA + k0);
        acc0 = __builtin_amdgcn_wmma_f32_16x16x4_f32(false, a0, false, b, (short)0, acc0, false, false);
        acc1 = __builtin_amdgcn_wmma_f32_16x16x4_f32(false, a1, false, b, (short)0, acc1, false, false);
        acc2 = __builtin_amdgcn_wmma_f32_16x16x4_f32(false, a2, false, b, (short)0, acc2, false, false);
        acc3 = __builtin_amdgcn_wmma_f32_16x16x4_f32(false, a3, false, b, (short)0, acc3, false, false);
    }

    const float bv = bias ? bias[n] : 0.0f;
    // C/D layout: col = n; row(t, v) = mblk*64 + t*16 + v + 8*half
    const int row0 = mblk * 64 + 8 * half;
    float*       Dbase  = D + (size_t)row0 * NOUT + n;
    float*       Gbase  = AGG ? (AGG + (size_t)row0 * NOUT + n) : nullptr;
    const float* dibase = dinv + row0;

    auto store_tile = [&](int t, const v8f& cc) {
        #pragma unroll
        for (int v = 0; v < 8; ++v) {
            const int ro = t * 16 + v;                // relative row (const)
            const float val = cc[v] + bv;
            Dbase[(size_t)ro * NOUT] = val;
            if (AGG) {
                const float di = dibase[ro];
                Gbase[(size_t)ro * NOUT] = val * di * di;
            }
        }
    };
    store_tile(0, acc0);
    store_tile(1, acc1);
    store_tile(2, acc2);
    store_tile(3, acc3);
}

// -------------------------------------------------- edge scatter (atomics)
// agg[dst] += h[src] * dinv[src]*dinv[dst] ; one wave per edge, float4/lane
__global__ __launch_bounds__(256)
void k_edge_scatter(const float* __restrict__ h, float* __restrict__ agg,
                    const int* __restrict__ src, const int* __restrict__ dst,
                    const float* __restrict__ dinv) {
    const int e = blockIdx.x * 8 + (threadIdx.x >> 5);
    if (e >= NEDGES) return;
    const int lane = threadIdx.x & 31;
    const int s = src[e], d = dst[e];
    const float w = dinv[s] * dinv[d];
    const float4 hv = *reinterpret_cast<const float4*>(h + (size_t)s * HID + lane * 4);
    float* out = agg + (size_t)d * HID + lane * 4;
    atomicAddF(out + 0, hv.x * w);
    atomicAddF(out + 1, hv.y * w);
    atomicAddF(out + 2, hv.z * w);
    atomicAddF(out + 3, hv.w * w);
}

// -------------------------------------------------------------- bias + act
__global__ void k_bias_act(float* __restrict__ x, const float* __restrict__ b,
                           int do_relu) {
    const int i = blockIdx.x;          // node (< NNODES)
    const int c = threadIdx.x;         // channel
    float v = x[(size_t)i * HID + c] + b[c];
    if (do_relu) v = fmaxf(v, 0.0f);
    x[(size_t)i * HID + c] = v;
}

// ------------------------------------------------------------------ pooling
__global__ void k_pool_accum(const float* __restrict__ act,
                             const int* __restrict__ batch,
                             float* __restrict__ pooled, float* __restrict__ cnt) {
    const int i = blockIdx.x;
    const int c = threadIdx.x;
    const int g = batch[i];
    atomicAddF(&pooled[(size_t)g * HID + c], act[(size_t)i * HID + c]);
    if (c == 0) atomicAddF(&cnt[g], 1.0f);
}

__global__ void k_pool_div(float* __restrict__ pooled, const float* __restrict__ cnt) {
    const int g = blockIdx.x;
    const int c = threadIdx.x;
    pooled[(size_t)g * HID + c] /= fmaxf(cnt[g], 1.0f);
}

// --------------------------------------------- BatchNorm(256) + ReLU, G=64
__global__ void k_bn_relu(float* __restrict__ z) {
    const int j = threadIdx.x;         // column, 256 threads
    float mu = 0.0f;
    for (int g = 0; g < NG; ++g) mu += z[g * 256 + j];
    mu *= (1.0f / NG);
    float var = 0.0f;
    for (int g = 0; g < NG; ++g) {
        const float d = z[g * 256 + j] - mu;
        var += d * d;
    }
    var *= (1.0f / NG);                 // biased variance (jnp.var default)
    const float s = rsqrtf(var + BN_EPS);
    for (int g = 0; g < NG; ++g) {
        float v = (z[g * 256 + j] - mu) * s;
        z[g * 256 + j] = fmaxf(v, 0.0f);
    }
}

// ----------------------------------------------------- final linear 256->2
__global__ void k_final(const float* __restrict__ z, const float* __restrict__ W,
                        const float* __restrict__ b, float* __restrict__ out) {
    const int idx = threadIdx.x;       // 128 threads: (g, j)
    const int g = idx >> 1, j = idx & 1;
    float acc = b[j];
    for (int k = 0; k < 256; ++k) acc += z[g * 256 + k] * W[k * 2 + j];
    out[g * 2 + j] = acc;
}

// ---------------------------------------------------------------- launcher
extern "C" void kernel_launch(void* const* d_in, const int* in_sizes, int n_in,
                              void* d_out, int out_size, void* d_ws, size_t ws_size,
                              hipStream_t stream) {
    const float* x    = (const float*)d_in[0];
    const int*   ei   = (const int*)  d_in[1];   // [2,E]: src row then dst row
    /* d_in[2] edge_weight: unused by reference */
    const int*   batch= (const int*)  d_in[3];
    const float* W1 = (const float*)d_in[4];  const float* b1 = (const float*)d_in[5];
    const float* W2 = (const float*)d_in[6];  const float* b2 = (const float*)d_in[7];
    const float* W3 = (const float*)d_in[8];  const float* b3 = (const float*)d_in[9];
    const float* W4 = (const float*)d_in[10]; const float* b4 = (const float*)d_in[11];
    const float* Wl = (const float*)d_in[12]; const float* bl = (const float*)d_in[13];
    const float* Wl2= (const float*)d_in[14]; const float* bl2= (const float*)d_in[15];

    const int* srcv = ei;
    const int* dstv = ei + NEDGES;

    float* ws     = (float*)d_ws;
    float* dinv   = ws;                                   // NPAD
    float* xpad   = dinv + NPAD;                          // NPAD*16
    float* w1p    = xpad + (size_t)NPAD * 16;             // 16*HID
    float* hlin   = w1p  + 16 * HID;                      // NPAD*HID
    float* act0   = hlin + (size_t)NPAD * HID;            // NPAD*HID
    float* act1   = act0 + (size_t)NPAD * HID;            // NPAD*HID
    float* pooled = act1 + (size_t)NPAD * HID;            // NG*HID
    float* cnt    = pooled + NG * HID;                    // NG
    float* z      = cnt + NG;                             // NG*256
    float* outp   = (float*)d_out;                        // [NG,2]

    // degrees (self-loop contributes 1), then dinv = rsqrt(deg); pad entries = 1
    k_fill<<<(NPAD + 255) / 256, 256, 0, stream>>>(dinv, 1.0f, NPAD);
    k_deg_accum<<<(NEDGES + 255) / 256, 256, 0, stream>>>(dstv, dinv);
    k_deg_to_dinv<<<(NPAD + 255) / 256, 256, 0, stream>>>(dinv, NPAD);

    // pack layer-1 operands (zero-padded K: 10 -> 16)
    k_pack_x<<<(NPAD * 16 + 255) / 256, 256, 0, stream>>>(x, xpad);
    k_pack_w1<<<(16 * HID + 255) / 256, 256, 0, stream>>>(W1, w1p);

    const int mblks   = NPAD / 64;                        // 782
    const int gblocks = mblks * (HID / 16) / 8;           // waves/8, exact
    const int eblocks = (NEDGES + 7) / 8;

    // layer 1: xpad @ W1p -> hlin ; self-loop init in act0 ; scatter ; bias+relu
    k_gemm_wmma<16, HID, 16><<<gblocks, 256, 0, stream>>>(xpad, w1p, hlin, nullptr, dinv, act0);
    k_edge_scatter<<<eblocks, 256, 0, stream>>>(hlin, act0, srcv, dstv, dinv);
    k_bias_act<<<NNODES, HID, 0, stream>>>(act0, b1, 1);

    // layer 2
    k_gemm_wmma<HID, HID, HID><<<gblocks, 256, 0, stream>>>(act0, W2, hlin, nullptr, dinv, act1);
    k_edge_scatter<<<eblocks, 256, 0, stream>>>(hlin, act1, srcv, dstv, dinv);
    k_bias_act<<<NNODES, HID, 0, stream>>>(act1, b2, 1);

    // layer 3
    k_gemm_wmma<HID, HID, HID><<<gblocks, 256, 0, stream>>>(act1, W3, hlin, nullptr, dinv, act0);
    k_edge_scatter<<<eblocks, 256, 0, stream>>>(hlin, act0, srcv, dstv, dinv);
    k_bias_act<<<NNODES, HID, 0, stream>>>(act0, b3, 1);

    // layer 4 (no relu)
    k_gemm_wmma<HID, HID, HID><<<gblocks, 256, 0, stream>>>(act0, W4, hlin, nullptr, dinv, act1);
    k_edge_scatter<<<eblocks, 256, 0, stream>>>(hlin, act1, srcv, dstv, dinv);
    k_bias_act<<<NNODES, HID, 0, stream>>>(act1, b4, 0);

    // global mean pool (pooled and cnt are contiguous: zero both in one pass)
    k_fill<<<(NG * HID + NG + 255) / 256, 256, 0, stream>>>(pooled, 0.0f, NG * HID + NG);
    k_pool_accum<<<NNODES, HID, 0, stream>>>(act1, batch, pooled, cnt);
    k_pool_div<<<NG, HID, 0, stream>>>(pooled, cnt);

    // head: z = pooled @ Wl + bl  (M=64 -> exactly 16 waves = 2 blocks)
    k_gemm_wmma<HID, 256, HID><<<2, 256, 0, stream>>>(pooled, Wl, z, bl, nullptr, nullptr);
    k_bn_relu<<<1, 256, 0, stream>>>(z);
    k_final<<<1, 128, 0, stream>>>(z, Wl2, bl2, outp);
}